// FeatTrans1d_4612794876088
// MI455X (gfx1250) — compile-verified
//
#include <hip/hip_runtime.h>

// FeatTrans1d on MI455X (gfx1250): fp32 WMMA 16x16x4 path.
// N=256, M=32, C=4, F=256, OUT = 4*256 = 1024.
//
// Per block (one n): GEMM1 H(33x256) x adj_c(256x256) per channel (adj built
// on the fly in LDS, symmetric), then GEMM2 H2(33x1024) x W^T(1024x1024).
// ~35 GFLOP fp32 vs ~100 MB HBM -> matrix-compute bound -> fp32 WMMA.

typedef __attribute__((ext_vector_type(2))) float v2f;
typedef __attribute__((ext_vector_type(8))) float v8f;

#define NPTS 256
#define MNB  32
#define CCH  4
#define FDIM 256
#define ODIM 1024   // OUT_C * OUT_F
#define ROWP 260    // padded row length (floats) for nb / H2
#define BPAD 18     // padded row length for adjT

#if __has_builtin(__builtin_amdgcn_global_load_async_to_lds_b128) && \
    __has_builtin(__builtin_amdgcn_s_wait_asynccnt)
#define USE_ASYNC_LDS 1
#else
#define USE_ASYNC_LDS 0
#endif

#if USE_ASYNC_LDS
// Builtin signature (from compiler diagnostic): arg0 is
//   'int __attribute__((vector_size(16))) __device__ *'  (addrspace(1) v4i*)
typedef int v4i_n __attribute__((vector_size(16)));
typedef __attribute__((address_space(1))) v4i_n gv4i;
typedef __attribute__((address_space(3))) v4i_n lv4i;

// Cross-address-space casts via integers (always legal in clang):
//  - global: generic and AS1 addresses are numerically identical
//  - LDS: low 32 bits of a generic LDS address are the LDS offset
__device__ __forceinline__ gv4i* to_gas(const void* p) {
    return (gv4i*)(unsigned long long)p;
}
__device__ __forceinline__ lv4i* to_lds(void* p) {
    return (lv4i*)(unsigned int)(unsigned long long)p;
}
#endif

__launch_bounds__(512, 1)
__global__ void feattrans_wmma_kernel(const float* __restrict__ x,
                                      const float* __restrict__ nbr,
                                      const float* __restrict__ W,
                                      float* __restrict__ out)
{
    // ---- LDS (~308 KB total, <= 320 KB/WGP) ----
    __shared__ float xs[CCH * FDIM];           //   4 KB  x[n,c,f]
    __shared__ float ss[CCH * FDIM];           //   4 KB  S[n,c,f] = sum_m neighbor
    __shared__ float nb[CCH][MNB][ROWP];       // 130 KB  neighbor[n,m,c,f] -> [c][m][f]
    __shared__ float adjT[2][FDIM][BPAD];      //  36 KB  adj chunk ping-pong, [a][b%16]
    __shared__ float H2[CCH][33][ROWP];        // 134 KB  rows 0-31 = nt, row 32 = xt

    const int n    = blockIdx.x;
    const int tid  = threadIdx.x;
    const int lane = tid & 31;          // wave32
    const int wv   = tid >> 5;          // wave id 0..15
    const int lh   = (lane >> 4) & 1;   // 0 = lanes 0-15, 1 = lanes 16-31
    const int l16  = lane & 15;

    // ---- stage neighbor[n] into nb[c][m][f] ----
    const float* nbase = nbr + (size_t)n * (MNB * CCH * FDIM);
#if USE_ASYNC_LDS
    for (int i = tid * 4; i < MNB * CCH * FDIM; i += 512 * 4) {
        int m = i >> 10;          // c*f span = 1024
        int c = (i >> 8) & 3;
        int f = i & 255;
        __builtin_amdgcn_global_load_async_to_lds_b128(
            to_gas(nbase + i), to_lds(&nb[c][m][f]), 0, 0);
    }
#else
    for (int i = tid * 4; i < MNB * CCH * FDIM; i += 512 * 4) {
        float4 v = *(const float4*)(nbase + i);
        int m = i >> 10;
        int c = (i >> 8) & 3;
        int f = i & 255;
        *(float4*)&nb[c][m][f] = v;
    }
#endif

    // ---- stage x[n] (overlaps with async neighbor staging) ----
    for (int i = tid; i < CCH * FDIM; i += 512)
        xs[i] = x[n * CCH * FDIM + i];

#if USE_ASYNC_LDS
    __builtin_amdgcn_s_wait_asynccnt(0);
#endif
    __syncthreads();

    // ---- S[c][f] = sum_m neighbor[n,m,c,f] ----
    for (int i = tid; i < CCH * FDIM; i += 512) {
        int c = i >> 8, f = i & 255;
        float s = 0.f;
        #pragma unroll 8
        for (int m = 0; m < MNB; ++m) s += nb[c][m][f];
        ss[i] = s;
    }
    __syncthreads();   // ss visible to all waves before adj compute

    // =========================================================================
    // GEMM1, per channel c:  H2[c] (33 x 256) = H (rows: 32 nbr + x) x adj[c]
    //   adj[c][a][b] = sgnroot(x_a*S_b + x_b*S_a) / (sum_c' |sgnroot| + 1e-7)
    //   adj symmetric -> B-operand adj[b][a] == adjT[a][b].
    // Ping-pong adjT buffers: one barrier per b-chunk.
    // Wave w owns output column tile a in [16w, 16w+16) for all 3 row tiles.
    // =========================================================================
    for (int c = 0; c < CCH; ++c) {
        v8f acc0, acc1, acc2;
        #pragma unroll
        for (int j = 0; j < 8; ++j) { acc0[j] = 0.f; acc1[j] = 0.f; acc2[j] = 0.f; }

        for (int bc = 0; bc < 16; ++bc) {
            const int p = bc & 1;
            // compute adj chunk for b in [16*bc, 16*bc+16), all a, channel c
            for (int e = tid; e < FDIM * 16; e += 512) {
                int a  = e >> 4;
                int bi = e & 15;
                int b  = bc * 16 + bi;
                float denom = 0.f, tc = 0.f;
                #pragma unroll
                for (int c2 = 0; c2 < CCH; ++c2) {
                    float u  = xs[c2 * FDIM + a] * ss[c2 * FDIM + b]
                             + xs[c2 * FDIM + b] * ss[c2 * FDIM + a];
                    float r  = __builtin_sqrtf(fmaxf(fabsf(u), 1e-8f));
                    float sr = (u == 0.f) ? 0.f : __builtin_copysignf(r, u);  // sign(0)=0
                    denom += fabsf(sr);
                    if (c2 == c) tc = sr;
                }
                adjT[p][a][bi] = tc / (denom + 1e-7f);
            }
            __syncthreads();

            #pragma unroll
            for (int kc = 0; kc < 4; ++kc) {
                // B 4x16: VGPR0 = row K+0 (lanes 0-15) / K+2 (lanes 16-31); VGPR1 = K+1 / K+3
                int r0  = kc * 4 + 2 * lh;
                v2f bop = *(const v2f*)&adjT[p][wv * 16 + l16][r0];
                // A 16x4: VGPR0 = K+0 / K+2, VGPR1 = K+1 / K+3; M = lane%16
                int kb  = bc * 16 + kc * 4 + 2 * lh;
                v2f a0  = *(const v2f*)&nb[c][l16][kb];        // rows 0-15
                v2f a1  = *(const v2f*)&nb[c][16 + l16][kb];   // rows 16-31
                v2f a2;                                        // row 32 = x, rows 33-47 = 0
                if (l16 == 0) a2 = *(const v2f*)&xs[c * FDIM + kb];
                else { a2.x = 0.f; a2.y = 0.f; }

                acc0 = __builtin_amdgcn_wmma_f32_16x16x4_f32(false, a0, false, bop, (short)0, acc0, false, false);
                acc1 = __builtin_amdgcn_wmma_f32_16x16x4_f32(false, a1, false, bop, (short)0, acc1, false, false);
                acc2 = __builtin_amdgcn_wmma_f32_16x16x4_f32(false, a2, false, bop, (short)0, acc2, false, false);
            }
        }
        // C/D layout: VGPR j -> M = j (lanes 0-15) / j+8 (lanes 16-31), N = lane%16
        #pragma unroll
        for (int j = 0; j < 8; ++j) {
            int rlo = j + 8 * lh;
            H2[c][rlo][wv * 16 + l16]      = acc0[j];
            H2[c][16 + rlo][wv * 16 + l16] = acc1[j];
            if (rlo == 0) H2[c][32][wv * 16 + l16] = acc2[j];   // row 32 = xt
        }
    }
    __syncthreads();

    // =========================================================================
    // GEMM2: out (33 x 1024) = H2 (33 x K=1024 over (c,f)) x W^T (1024 x 1024)
    //   B[k][o] = W[o*1024 + k]  (streamed from global; W is L2-resident, 4 MB;
    //   the 128B L0 lines of W are reused across 8 consecutive k-steps)
    // Wave w owns output columns ot in {w, w+16, w+32, w+48}.
    // =========================================================================
    float* outx = out + (size_t)n * ODIM;                                  // x_out (N,1024)
    float* outn = out + (size_t)NPTS * ODIM + (size_t)n * MNB * ODIM;      // n_out (N,M,1024)

    for (int g = 0; g < 4; ++g) {
        const int ot = g * 16 + wv;       // 0..63
        const int o  = ot * 16 + l16;
        v8f acc0, acc1, acc2;
        #pragma unroll
        for (int j = 0; j < 8; ++j) { acc0[j] = 0.f; acc1[j] = 0.f; acc2[j] = 0.f; }

        #pragma unroll 4
        for (int k = 0; k < ODIM; k += 4) {
            int kb  = k + 2 * lh;
            v2f bop = *(const v2f*)&W[(size_t)o * ODIM + kb];  // {W[o][kb], W[o][kb+1]}
            int c2  = kb >> 8;
            int f   = kb & 255;
            v2f a0  = *(const v2f*)&H2[c2][l16][f];
            v2f a1  = *(const v2f*)&H2[c2][16 + l16][f];
            v2f a2;
            if (l16 == 0) a2 = *(const v2f*)&H2[c2][32][f];
            else { a2.x = 0.f; a2.y = 0.f; }

            acc0 = __builtin_amdgcn_wmma_f32_16x16x4_f32(false, a0, false, bop, (short)0, acc0, false, false);
            acc1 = __builtin_amdgcn_wmma_f32_16x16x4_f32(false, a1, false, bop, (short)0, acc1, false, false);
            acc2 = __builtin_amdgcn_wmma_f32_16x16x4_f32(false, a2, false, bop, (short)0, acc2, false, false);
        }

        #pragma unroll
        for (int j = 0; j < 8; ++j) {
            int rlo = j + 8 * lh;
            outn[(size_t)rlo * ODIM + o]        = acc0[j];     // m = 0..15
            outn[(size_t)(16 + rlo) * ODIM + o] = acc1[j];     // m = 16..31
            if (rlo == 0) outx[o] = acc2[j];                   // row 32 -> x_out
        }
    }
}

extern "C" void kernel_launch(void* const* d_in, const int* in_sizes, int n_in,
                              void* d_out, int out_size, void* d_ws, size_t ws_size,
                              hipStream_t stream) {
    (void)in_sizes; (void)n_in; (void)out_size; (void)d_ws; (void)ws_size;
    const float* x   = (const float*)d_in[0];
    const float* nbr = (const float*)d_in[1];
    const float* W   = (const float*)d_in[2];
    feattrans_wmma_kernel<<<dim3(NPTS), dim3(512), 0, stream>>>(x, nbr, W, (float*)d_out);
}